// SADSimilarity_37383395344618
// MI455X (gfx1250) — compile-verified
//
#include <hip/hip_runtime.h>

// SAD similarity: out[b,n,m] = -sum_d |lhs[b,n,d] - rhs[b,m,d]|
// B=4, N=M=1024, D=64 (fp32). VALU-bound; WMMA is inapplicable (|a-b| not bilinear).
// CDNA5 path: global_load_async_to_lds_b128 + s_wait_asynccnt for tile staging.

#define TILE       64
#define DDIM       64
#define LDS_STRIDE 68   // floats per LDS row (272 B): 16B-aligned, bank-conflict-free reads

__global__ __launch_bounds__(256)
void sad_similarity_kernel(const float* __restrict__ lhs,
                           const float* __restrict__ rhs,
                           float* __restrict__ out,
                           int N, int M)
{
    __shared__ float lA[TILE * LDS_STRIDE];   // 17408 B
    __shared__ float lB[TILE * LDS_STRIDE];   // 17408 B

    const int tid = threadIdx.x;
    const int b   = blockIdx.z;
    const int n0  = blockIdx.y * TILE;
    const int m0  = blockIdx.x * TILE;

    // Each tile is a contiguous 64 rows x 256 B = 16 KB block in global memory
    // (row stride == D == 64 floats), i.e. 1024 16-byte chunks.
    const float* gA = lhs + ((size_t)b * N + n0) * DDIM;
    const float* gB = rhs + ((size_t)b * M + m0) * DDIM;

    // Generic pointer to __shared__ : low 32 bits == wave-relative LDS byte address.
    const uint32_t aBase = (uint32_t)(uintptr_t)(&lA[0]);
    const uint32_t bBase = (uint32_t)(uintptr_t)(&lB[0]);

    // Async copy: 4 chunks per thread per tile (256 threads x 4 = 1024 chunks).
    #pragma unroll
    for (int k = 0; k < 4; ++k) {
        const int chunk = tid + k * 256;        // 0..1023
        const int row   = chunk >> 4;           // 16 chunks per 256B row
        const int col   = chunk & 15;
        const uint32_t ldsOff = (uint32_t)(row * (LDS_STRIDE * 4) + col * 16);
        const uint64_t ga = (uint64_t)(uintptr_t)gA + (uint64_t)chunk * 16u;
        const uint64_t gb = (uint64_t)(uintptr_t)gB + (uint64_t)chunk * 16u;
        const uint32_t la = aBase + ldsOff;
        const uint32_t lb = bBase + ldsOff;
        asm volatile("global_load_async_to_lds_b128 %0, %1, off"
                     :: "v"(la), "v"(ga) : "memory");
        asm volatile("global_load_async_to_lds_b128 %0, %1, off"
                     :: "v"(lb), "v"(gb) : "memory");
    }
    asm volatile("s_wait_asynccnt 0x0" ::: "memory");
    __syncthreads();

    // Interleaved 4x4 micro-tile per thread: n = ty + 16*i, m = tx + 16*j.
    // rhs reads: lanes 0..15 -> byte offsets 272*tx (mod 256 = 16*tx): disjoint
    // bank quads covering all 64 banks -> conflict-free ds_load_b128.
    const int tx = tid & 15;
    const int ty = tid >> 4;

    float acc[4][4];
    #pragma unroll
    for (int i = 0; i < 4; ++i)
        #pragma unroll
        for (int j = 0; j < 4; ++j)
            acc[i][j] = 0.0f;

    #pragma unroll
    for (int d = 0; d < DDIM; d += 4) {
        float4 a4[4], b4[4];
        #pragma unroll
        for (int i = 0; i < 4; ++i)
            a4[i] = *reinterpret_cast<const float4*>(&lA[(ty + 16 * i) * LDS_STRIDE + d]);
        #pragma unroll
        for (int j = 0; j < 4; ++j)
            b4[j] = *reinterpret_cast<const float4*>(&lB[(tx + 16 * j) * LDS_STRIDE + d]);

        #pragma unroll
        for (int i = 0; i < 4; ++i) {
            #pragma unroll
            for (int j = 0; j < 4; ++j) {
                // v_sub_f32 + v_add_f32 with |src| modifier; VOPD dual-issue friendly
                acc[i][j] += __builtin_fabsf(a4[i].x - b4[j].x);
                acc[i][j] += __builtin_fabsf(a4[i].y - b4[j].y);
                acc[i][j] += __builtin_fabsf(a4[i].z - b4[j].z);
                acc[i][j] += __builtin_fabsf(a4[i].w - b4[j].w);
            }
        }
    }

    // Store: for fixed (i,j), lanes 0..15 write 64 consecutive bytes.
    #pragma unroll
    for (int i = 0; i < 4; ++i) {
        const int n = n0 + ty + 16 * i;
        float* orow = out + ((size_t)b * N + n) * M + m0;
        #pragma unroll
        for (int j = 0; j < 4; ++j) {
            orow[tx + 16 * j] = -acc[i][j];
        }
    }
}

extern "C" void kernel_launch(void* const* d_in, const int* in_sizes, int n_in,
                              void* d_out, int out_size, void* d_ws, size_t ws_size,
                              hipStream_t stream) {
    const float* lhs = (const float*)d_in[0];
    const float* rhs = (const float*)d_in[1];
    float* out = (float*)d_out;

    const int B = 4, D = 64;
    const int N = in_sizes[0] / (B * D);   // 1024
    const int M = in_sizes[1] / (B * D);   // 1024

    dim3 grid(M / TILE, N / TILE, B);      // 16 x 16 x 4
    dim3 block(256);
    sad_similarity_kernel<<<grid, block, 0, stream>>>(lhs, rhs, out, N, M);
}